// Temporal_Attention_46780783788082
// MI455X (gfx1250) — compile-verified
//
#include <hip/hip_runtime.h>
#include <math.h>

typedef __attribute__((ext_vector_type(2))) float v2f;
typedef __attribute__((ext_vector_type(8))) float v8f;

#define B_  16
#define T_  24
#define N_  2048
#define F_  64
#define NF_ (N_ * F_)      // 131072 columns per (b, t) row
#define TT_ (T_ * T_)      // 576
#define SP_ 32             // padded s-dimension for LDS W tile

// ---------------------------------------------------------------------------
// Kernel 1: one block per (b,t). Single streaming pass over x[b,t,:,:]:
//   rhs[b,t,n] = sum_f x[b,t,n,f] * U3[f]
//   v[f]       = sum_n x[b,t,n,f] * U1[n]      (64 per-thread accumulators)
// then lhs[b,t,n] = sum_f v[f] * U2[n,f].
// ---------------------------------------------------------------------------
__global__ void k_lhs_rhs(const float* __restrict__ x,
                          const float* __restrict__ U1,
                          const float* __restrict__ U2,
                          const float* __restrict__ U3,
                          float* __restrict__ lhs,
                          float* __restrict__ rhs) {
    __shared__ float red[256 * 65];   // padded stride 65 -> conflict-free
    __shared__ float v[F_];

    const int tid = threadIdx.x;
    const int bt  = blockIdx.x;                 // 0 .. B*T-1
    const size_t xbase = (size_t)bt * NF_;

    float4 vacc[16];
#pragma unroll
    for (int c = 0; c < 16; ++c) vacc[c] = make_float4(0.f, 0.f, 0.f, 0.f);

    const float4* U3v = (const float4*)U3;

    for (int n = tid; n < N_; n += 256) {
        const float4* row = (const float4*)(x + xbase + (size_t)n * F_);
        const float u1 = U1[n];
        float racc = 0.f;
#pragma unroll
        for (int c = 0; c < 16; ++c) {
            float4 xv = row[c];
            float4 u3 = U3v[c];
            racc += xv.x * u3.x + xv.y * u3.y + xv.z * u3.z + xv.w * u3.w;
            vacc[c].x += xv.x * u1;
            vacc[c].y += xv.y * u1;
            vacc[c].z += xv.z * u1;
            vacc[c].w += xv.w * u1;
        }
        rhs[(size_t)bt * N_ + n] = racc;
    }

    // dump per-thread v accumulators to LDS and reduce across 256 threads
#pragma unroll
    for (int c = 0; c < 16; ++c) {
        red[tid * 65 + 4 * c + 0] = vacc[c].x;
        red[tid * 65 + 4 * c + 1] = vacc[c].y;
        red[tid * 65 + 4 * c + 2] = vacc[c].z;
        red[tid * 65 + 4 * c + 3] = vacc[c].w;
    }
    __syncthreads();
    if (tid < F_) {
        float s = 0.f;
        for (int i = 0; i < 256; ++i) s += red[i * 65 + tid];
        v[tid] = s;
    }
    __syncthreads();

    // lhs[n] = sum_f v[f] * U2[n,f]
    for (int n = tid; n < N_; n += 256) {
        const float4* u2row = (const float4*)(U2 + (size_t)n * F_);
        float acc = 0.f;
#pragma unroll
        for (int c = 0; c < 16; ++c) {
            float4 u2 = u2row[c];
            acc += u2.x * v[4 * c + 0] + u2.y * v[4 * c + 1] +
                   u2.z * v[4 * c + 2] + u2.w * v[4 * c + 3];
        }
        lhs[(size_t)bt * N_ + n] = acc;
    }
}

// ---------------------------------------------------------------------------
// Kernel 2: one block per batch b. Tiny 24x24 chain:
//   S[i,j] = sigmoid( sum_n lhs[b,i,n]*rhs[b,j,n] + be[i,j] )
//   E[t,r] = sum_s Ve[t,s] * S[s,r]
//   A[b,t,r] = softmax over t of E[t,r]
// ---------------------------------------------------------------------------
__global__ void k_attn(const float* __restrict__ lhs,
                       const float* __restrict__ rhs,
                       const float* __restrict__ be,
                       const float* __restrict__ Ve,
                       float* __restrict__ Amat) {
    __shared__ float sp[TT_];
    __shared__ float se[TT_];

    const int tid = threadIdx.x;
    const int b   = blockIdx.x;

    for (int p = tid; p < TT_; p += 256) {
        const int t = p / T_, s = p % T_;
        const float4* L = (const float4*)(lhs + (size_t)(b * T_ + t) * N_);
        const float4* R = (const float4*)(rhs + (size_t)(b * T_ + s) * N_);
        float acc = 0.f;
        for (int n = 0; n < N_ / 4; ++n) {
            float4 lv = L[n];
            float4 rv = R[n];
            acc += lv.x * rv.x + lv.y * rv.y + lv.z * rv.z + lv.w * rv.w;
        }
        const float z = acc + be[t * T_ + s];
        sp[p] = 1.0f / (1.0f + expf(-z));
    }
    __syncthreads();

    for (int p = tid; p < TT_; p += 256) {
        const int t = p / T_, r = p % T_;
        float e = 0.f;
#pragma unroll
        for (int s = 0; s < T_; ++s) e += Ve[t * T_ + s] * sp[s * T_ + r];
        se[p] = e;
    }
    __syncthreads();

    if (tid < T_) {
        const int r = tid;
        float mx = -1e30f;
#pragma unroll
        for (int t = 0; t < T_; ++t) mx = fmaxf(mx, se[t * T_ + r]);
        float sum = 0.f;
        float ex[T_];
#pragma unroll
        for (int t = 0; t < T_; ++t) { ex[t] = expf(se[t * T_ + r] - mx); sum += ex[t]; }
        const float inv = 1.0f / sum;
#pragma unroll
        for (int t = 0; t < T_; ++t) Amat[(size_t)b * TT_ + t * T_ + r] = ex[t] * inv;
    }
}

// ---------------------------------------------------------------------------
// Kernel 3 (WMMA): out[b,s,c] = sum_t W[s,t] * x[b,t,c],  W[s,t] = A[b,t,s]
// per batch: M = 24 (s, two 16-row tiles, zero-padded), K = 24 (t, six K=4
// chunks), N = 131072 columns. One wave per 16-column tile; 8 waves / block.
//
// The 24x24 attention matrix is staged once per block into LDS, zero-padded
// to Wp[24][32] (t-major), so every A-operand fetch is an unconditional
// ds_load (lowers to ds_load_2addr_b32 pairs) — no exec-mask branching in
// the WMMA inner loop. The s=16..23 stores of the second tile live under a
// single hi==0 divergence region (one saveexec instead of eight).
//
// V_WMMA_F32_16X16X4_F32 operand layouts per CDNA5 ISA 7.12.2:
//   A 16x4: lanes 0-15 -> M=lane, {K=k0, K=k0+1}; lanes 16-31 -> {K=k0+2, k0+3}
//   B 4x16: lanes 0-15 -> col=lane, rows {k0, k0+1}; lanes 16-31 rows {k0+2, k0+3}
//   C/D 16x16: VGPR j: lanes 0-15 -> M=j, lanes 16-31 -> M=j+8
// ---------------------------------------------------------------------------
__global__ void k_out_wmma(const float* __restrict__ x,
                           const float* __restrict__ Amat,
                           float* __restrict__ out) {
    __shared__ float Wp[T_ * SP_];   // Wp[t][s], s padded 24->32 with zeros

    const int tid  = threadIdx.x;
    const int wave = tid >> 5;
    const int lane = tid & 31;
    const int lo   = lane & 15;
    const int hi   = lane >> 4;

    const int b = blockIdx.y;

    // cooperative zero-padded stage of A[b] into LDS (768 = 3 * 256)
    {
        const float* Ab = Amat + (size_t)b * TT_;
#pragma unroll
        for (int k = 0; k < 3; ++k) {
            const int i = tid + k * 256;         // i = t*32 + s
            const int t = i >> 5;
            const int s = i & 31;
            Wp[i] = (s < T_) ? Ab[t * T_ + s] : 0.0f;
        }
    }
    __syncthreads();

    const int col = blockIdx.x * 128 + wave * 16 + lo;   // nf column
    const float* xb = x + (size_t)b * T_ * NF_ + col;    // row stride = NF_

    const int s0 = lo;         // rows 0..15  (tile m=0)
    const int s1 = 16 + lo;    // rows 16..31 (tile m=1; rows >=24 are zeros)

    v8f acc0 = {};
    v8f acc1 = {};

#pragma unroll
    for (int kc = 0; kc < 6; ++kc) {
        const int k0 = kc * 4 + 2 * hi;          // this half-wave's K pair

        // B operand: x rows k0, k0+1 at this column
        v2f Bv;
        Bv.x = xb[(size_t)k0 * NF_];
        Bv.y = xb[(size_t)(k0 + 1) * NF_];

        // A operand tile m=0: W[s0, k0..k0+1] = Wp[k0..k0+1][s0]
        v2f Av0;
        Av0.x = Wp[k0 * SP_ + s0];
        Av0.y = Wp[(k0 + 1) * SP_ + s0];

        // A operand tile m=1 (rows s>=24 read LDS zeros — no branching)
        v2f Av1;
        Av1.x = Wp[k0 * SP_ + s1];
        Av1.y = Wp[(k0 + 1) * SP_ + s1];

        acc0 = __builtin_amdgcn_wmma_f32_16x16x4_f32(
            false, Av0, false, Bv, (short)0, acc0, false, false);
        acc1 = __builtin_amdgcn_wmma_f32_16x16x4_f32(
            false, Av1, false, Bv, (short)0, acc1, false, false);
    }

    float* ob = out + (size_t)b * T_ * NF_ + col;

    // tile m=0: rows s = 0..15, all lanes store
#pragma unroll
    for (int j = 0; j < 8; ++j) {
        ob[(size_t)(j + 8 * hi) * NF_] = acc0[j];
    }

    // tile m=1: valid rows s = 16..23 are exactly the hi==0 half-wave
    // (s = 16 + j + 8*hi < 24  <=>  hi == 0) — single divergence region
    if (hi == 0) {
#pragma unroll
        for (int j = 0; j < 8; ++j) {
            ob[(size_t)(16 + j) * NF_] = acc1[j];
        }
    }
}

// ---------------------------------------------------------------------------
extern "C" void kernel_launch(void* const* d_in, const int* in_sizes, int n_in,
                              void* d_out, int out_size, void* d_ws, size_t ws_size,
                              hipStream_t stream) {
    (void)in_sizes; (void)n_in; (void)out_size; (void)ws_size;

    const float* x  = (const float*)d_in[0];
    const float* U1 = (const float*)d_in[1];
    const float* U2 = (const float*)d_in[2];
    const float* U3 = (const float*)d_in[3];
    const float* be = (const float*)d_in[4];
    const float* Ve = (const float*)d_in[5];
    float* out = (float*)d_out;

    float* lhs  = (float*)d_ws;                        // B*T*N floats (3 MB)
    float* rhs  = lhs + (size_t)B_ * T_ * N_;          // B*T*N floats (3 MB)
    float* Amat = rhs + (size_t)B_ * T_ * N_;          // B*T*T floats (36 KB)

    hipLaunchKernelGGL(k_lhs_rhs, dim3(B_ * T_), dim3(256), 0, stream,
                       x, U1, U2, U3, lhs, rhs);
    hipLaunchKernelGGL(k_attn, dim3(B_), dim3(256), 0, stream,
                       lhs, rhs, be, Ve, Amat);
    hipLaunchKernelGGL(k_out_wmma, dim3(NF_ / 128, B_), dim3(256), 0, stream,
                       x, Amat, out);
}